// GCNNet_61950608278028
// MI455X (gfx1250) — compile-verified
//
#include <hip/hip_runtime.h>
#include <hip/hip_bf16.h>

typedef __attribute__((ext_vector_type(2))) float v2f;
typedef __attribute__((ext_vector_type(8))) float v8f;

#define F_IN 128
#define HID  16
#define NCLS 40
#define NCLS_PAD 48

// ---------------------------------------------------------------------------
// Kernel 1: H1[N1,16] = x[N1,128] @ W1[128,16]   (f32 WMMA 16x16x4)
// One wave per 16-row tile; 4 waves (128 threads) per block.
// x tile staged via CDNA5 async global->LDS copies (ASYNCcnt-tracked),
// no VGPR round-trip.
// ---------------------------------------------------------------------------
__global__ __launch_bounds__(128) void gcn_gemm1(const float* __restrict__ x,
                                                 const float* __restrict__ W1,
                                                 float* __restrict__ H1) {
    __shared__ __align__(16) float sW[F_IN * HID];   // 8 KB
    __shared__ __align__(16) float sX[4][16 * F_IN]; // 32 KB (8 KB per wave)

    const int tid  = threadIdx.x;
    const int wave = tid >> 5;
    const int lane = tid & 31;

    // --- async stage of this wave's 16x128 f32 tile of x -------------------
    // lane L copies 16B chunks; INST_OFFSET advances global+LDS together.
    const long tile = (long)blockIdx.x * 4 + wave;   // N1/16 tiles, divisible
    const long r0   = tile * 16;
    {
        const unsigned long long gbase =
            (unsigned long long)(x + r0 * F_IN) + (unsigned)(lane * 16);
        const unsigned ldsA =
            (unsigned)(unsigned long long)&sX[wave][0] + (unsigned)(lane * 16);
#pragma unroll
        for (int i = 0; i < 16; ++i) {
            asm volatile("global_load_async_to_lds_b128 %0, %1, off offset:%2"
                         :
                         : "v"(ldsA), "v"(gbase), "i"(i * 512)
                         : "memory");
        }
    }

    // stage W1 (row-major [128][16]) cooperatively with plain stores
    for (int i = tid; i < F_IN * HID; i += 128) sW[i] = W1[i];

    // wait for this wave's async tile, then block barrier for sW
    asm volatile("s_wait_asynccnt 0x0" ::: "memory");
    __syncthreads();

    const int m = lane & 15;        // A row / D col index
    const int h = lane >> 4;        // lane-half selects K phase / D row-half
    const int n = m;                // B/D column for this lane

    v8f c = {};
#pragma unroll
    for (int k0 = 0; k0 < F_IN; k0 += 4) {
        const int k = k0 + 2 * h;
        v2f a; a.x = sX[wave][m * F_IN + k];  a.y = sX[wave][m * F_IN + k + 1];
        v2f b; b.x = sW[k * HID + n];         b.y = sW[(k + 1) * HID + n];
        c = __builtin_amdgcn_wmma_f32_16x16x4_f32(false, a, false, b,
                                                  (short)0, c, false, false);
    }

    // D layout: VGPR r -> row (r + 8h), col n
#pragma unroll
    for (int r = 0; r < 8; ++r)
        H1[(r0 + r + 8 * h) * HID + n] = c[r];
}

// ---------------------------------------------------------------------------
// Kernel 2: scatter-sum layer 1: out1[dst1[e], f] += H1[src1[e], f] * ew1[e]
// one lane per (edge, feature), f in [0,16)
// ---------------------------------------------------------------------------
__global__ __launch_bounds__(256) void gcn_scatter1(const float* __restrict__ H1,
                                                    const int* __restrict__ src,
                                                    const int* __restrict__ dst,
                                                    const float* __restrict__ ew,
                                                    float* __restrict__ out1,
                                                    long nE) {
    const long t = (long)blockIdx.x * 256 + threadIdx.x;
    const long e = t >> 4;
    const int  f = (int)(t & 15);
    if (e >= nE) return;
    const float v = H1[(long)src[e] * HID + f] * ew[e];
    unsafeAtomicAdd(&out1[(long)dst[e] * HID + f], v);
}

// ---------------------------------------------------------------------------
// Kernel 3: H3[N2,40] = relu(out1[N2,16] + b1) @ W2[16,40]   (f32 WMMA)
// N padded to 48 -> 3 column tiles of 16; K = 16 -> 4 WMMA steps.
// ---------------------------------------------------------------------------
__global__ __launch_bounds__(128) void gcn_gemm2(const float* __restrict__ out1,
                                                 const float* __restrict__ b1,
                                                 const float* __restrict__ W2,
                                                 float* __restrict__ H3,
                                                 long nTiles) {
    __shared__ float sW[HID * NCLS_PAD];    // 16 x 48, zero-padded (3 KB)

    const int tid  = threadIdx.x;
    const int wave = tid >> 5;
    const int lane = tid & 31;

    for (int i = tid; i < HID * NCLS_PAD; i += 128) {
        const int k = i / NCLS_PAD, nn = i % NCLS_PAD;
        sW[i] = (nn < NCLS) ? W2[k * NCLS + nn] : 0.0f;
    }
    __syncthreads();

    const long tile = (long)blockIdx.x * 4 + wave;
    if (tile >= nTiles) return;
    const long r0 = tile * 16;

    const int m = lane & 15;
    const int h = lane >> 4;

    // A row for this lane: relu(out1[r0+m, :] + b1)
    float areg[HID];
#pragma unroll
    for (int k = 0; k < HID; ++k) {
        const float v = out1[(r0 + m) * HID + k] + b1[k];
        areg[k] = v > 0.0f ? v : 0.0f;
    }

#pragma unroll
    for (int nt = 0; nt < 3; ++nt) {
        const int n = nt * 16 + m;
        v8f c = {};
#pragma unroll
        for (int k0 = 0; k0 < HID; k0 += 4) {
            const int k = k0 + 2 * h;
            v2f a; a.x = areg[k];                  a.y = areg[k + 1];
            v2f b; b.x = sW[k * NCLS_PAD + n];     b.y = sW[(k + 1) * NCLS_PAD + n];
            c = __builtin_amdgcn_wmma_f32_16x16x4_f32(false, a, false, b,
                                                      (short)0, c, false, false);
        }
        if (n < NCLS) {
#pragma unroll
            for (int r = 0; r < 8; ++r)
                H3[(r0 + r + 8 * h) * NCLS + n] = c[r];
        }
    }
}

// ---------------------------------------------------------------------------
// Kernel 4: scatter-sum layer 2 directly into d_out:
//   out[dst2[e], f] += H3[src2[e], f] * ew2[e],  f in [0,40)
// ---------------------------------------------------------------------------
__global__ __launch_bounds__(256) void gcn_scatter2(const float* __restrict__ H3,
                                                    const int* __restrict__ src,
                                                    const int* __restrict__ dst,
                                                    const float* __restrict__ ew,
                                                    float* __restrict__ out,
                                                    long nE) {
    const long t = (long)blockIdx.x * 256 + threadIdx.x;
    const long e = t / NCLS;
    const int  f = (int)(t % NCLS);
    if (e >= nE) return;
    const float v = H3[(long)src[e] * NCLS + f] * ew[e];
    unsafeAtomicAdd(&out[(long)dst[e] * NCLS + f], v);
}

// ---------------------------------------------------------------------------
// Kernel 5: out[r,:] = log_softmax(out[r,:] + b2), one thread per row
// ---------------------------------------------------------------------------
__global__ __launch_bounds__(256) void gcn_lsm(float* __restrict__ out,
                                               const float* __restrict__ b2,
                                               int nRows) {
    const int r = blockIdx.x * 256 + threadIdx.x;
    if (r >= nRows) return;
    float v[NCLS];
    float mx = -3.402823466e+38f;
#pragma unroll
    for (int c = 0; c < NCLS; ++c) {
        v[c] = out[(long)r * NCLS + c] + b2[c];
        mx = fmaxf(mx, v[c]);
    }
    float s = 0.0f;
#pragma unroll
    for (int c = 0; c < NCLS; ++c) s += expf(v[c] - mx);
    const float l = mx + logf(s);
#pragma unroll
    for (int c = 0; c < NCLS; ++c) out[(long)r * NCLS + c] = v[c] - l;
}

// ---------------------------------------------------------------------------
extern "C" void kernel_launch(void* const* d_in, const int* in_sizes, int n_in,
                              void* d_out, int out_size, void* d_ws, size_t ws_size,
                              hipStream_t stream) {
    const float* x    = (const float*)d_in[0];
    const int*   src1 = (const int*)  d_in[1];
    const int*   dst1 = (const int*)  d_in[2];
    const float* ew1  = (const float*)d_in[3];
    const int*   src2 = (const int*)  d_in[4];
    const int*   dst2 = (const int*)  d_in[5];
    const float* ew2  = (const float*)d_in[6];
    const float* W1   = (const float*)d_in[7];
    const float* b1   = (const float*)d_in[8];
    const float* W2   = (const float*)d_in[9];
    const float* b2   = (const float*)d_in[10];
    float*       out  = (float*)d_out;

    const long N1 = (long)in_sizes[0] / F_IN;   // 1,000,000
    const long E1 = (long)in_sizes[1];          // 5,000,000
    const long E2 = (long)in_sizes[4];          //   500,000
    const long N2 = 500000;
    const int  B  = out_size / NCLS;            //    50,000

    // workspace layout (bytes): H1 [N1*16*4] | out1 [N2*16*4] | H3 [N2*40*4]
    char* ws = (char*)d_ws;
    float* H1   = (float*)(ws);
    float* out1 = (float*)(ws + (size_t)N1 * HID * 4);
    float* H3   = (float*)(ws + (size_t)N1 * HID * 4 + (size_t)N2 * HID * 4);

    // zero the accumulators (graph-capture-legal async memsets)
    hipMemsetAsync(out1, 0, (size_t)N2 * HID * 4, stream);
    hipMemsetAsync(out,  0, (size_t)B * NCLS * 4, stream);

    // layer 1 GEMM: N1/16 tiles, 4 tiles per block
    {
        const long tiles = N1 / 16;
        gcn_gemm1<<<(int)(tiles / 4), 128, 0, stream>>>(x, W1, H1);
    }
    // layer 1 scatter
    {
        const long threads = E1 * HID;
        gcn_scatter1<<<(int)((threads + 255) / 256), 256, 0, stream>>>(
            H1, src1, dst1, ew1, out1, E1);
    }
    // layer 2 GEMM (fused relu + b1)
    {
        const long tiles = N2 / 16;
        gcn_gemm2<<<(int)((tiles + 3) / 4), 128, 0, stream>>>(out1, b1, W2, H3, tiles);
    }
    // layer 2 scatter into d_out
    {
        const long threads = E2 * NCLS;
        gcn_scatter2<<<(int)((threads + 255) / 256), 256, 0, stream>>>(
            H3, src2, dst2, ew2, out, E2);
    }
    // bias + log_softmax in place
    gcn_lsm<<<(B + 255) / 256, 256, 0, stream>>>(out, b2, B);
}